// Learnable_MMF_Smooth_Wavelets_43576738185285
// MI455X (gfx1250) — compile-verified
//
#include <hip/hip_runtime.h>

typedef float v2f __attribute__((ext_vector_type(2)));
typedef float v8f __attribute__((ext_vector_type(8)));

#define NDIM  2048
#define HALFN 1024

// ------------------------------------------------------------------
// Scan phase: per-layer 16x16 rotation of rows / columns.
// ------------------------------------------------------------------

// Blocks 0..7: A row update (column slices). Blocks 8..15: right row update.
__global__ __launch_bounds__(256) void layer_rows(float* __restrict__ A,
                                                  float* __restrict__ R,
                                                  const float* __restrict__ O,
                                                  const int* __restrict__ sel,
                                                  int l)
{
    __shared__ float Ol[256];
    __shared__ int   idx[16];
    const int t = threadIdx.x;
    Ol[t] = O[l * 256 + t];
    if (t < 16) idx[t] = sel[l * 16 + t];
    __syncthreads();

    float* M = (blockIdx.x < 8) ? A : R;
    const int c = (blockIdx.x & 7) * 256 + t;

    float v[16];
#pragma unroll
    for (int j = 0; j < 16; ++j) v[j] = M[(size_t)idx[j] * NDIM + c];
#pragma unroll
    for (int i = 0; i < 16; ++i) {
        float s = 0.0f;
#pragma unroll
        for (int j = 0; j < 16; ++j) s = fmaf(Ol[i * 16 + j], v[j], s);
        M[(size_t)idx[i] * NDIM + c] = s;
    }
}

// Column update A[:, idx] = A[:, idx] @ Ol^T, plus stash of the 4 drop rows
// into mother_wavelets. Block b owns rows [256b, 256b+256); the drop rows'
// updated entries were written by this same block, so after a fence+barrier
// the block can copy the fully-updated rows.
__global__ __launch_bounds__(256) void layer_cols(float* __restrict__ A,
                                                  float* __restrict__ mw,
                                                  const float* __restrict__ O,
                                                  const int* __restrict__ sel,
                                                  const int* __restrict__ drop,
                                                  int l)
{
    __shared__ float Ol[256];
    __shared__ int   idx[16];
    __shared__ int   didx[4];
    const int t = threadIdx.x;
    Ol[t] = O[l * 256 + t];
    if (t < 16) idx[t] = sel[l * 16 + t];
    if (t < 4)  didx[t] = drop[l * 4 + t];
    __syncthreads();

    const int r = blockIdx.x * 256 + t;
    float v[16];
#pragma unroll
    for (int j = 0; j < 16; ++j) v[j] = A[(size_t)r * NDIM + idx[j]];
#pragma unroll
    for (int i = 0; i < 16; ++i) {
        float s = 0.0f;
#pragma unroll
        for (int j = 0; j < 16; ++j) s = fmaf(v[j], Ol[i * 16 + j], s);
        A[(size_t)r * NDIM + idx[i]] = s;
    }

    __threadfence_block();
    __syncthreads();

#pragma unroll
    for (int d = 0; d < 4; ++d) {
        const int dr = didx[d];
        if ((dr >> 8) == (int)blockIdx.x) {
            for (int c = t; c < NDIM; c += 256)
                mw[((size_t)l * 4 + d) * NDIM + c] = A[(size_t)dr * NDIM + c];
        }
    }
}

// ------------------------------------------------------------------
// Epilogue helpers
// ------------------------------------------------------------------
__global__ void init_right(float* __restrict__ R)
{
    const size_t i = (size_t)blockIdx.x * blockDim.x + threadIdx.x;
    const int r = (int)(i >> 11), c = (int)(i & 2047);
    R[i] = (r == c) ? 1.0f : 0.0f;
}

// flags[i] = 1 iff i in sorted active_idx (branch-free binary search)
__global__ void make_flags(int* __restrict__ flags, const int* __restrict__ act)
{
    const int t = blockIdx.x * 256 + threadIdx.x;   // 0..2047
    int lo = 0, hi = HALFN;
    while (lo < hi) {
        const int mid = (lo + hi) >> 1;
        if (act[mid] < t) lo = mid + 1; else hi = mid;
    }
    flags[t] = (lo < HALFN && act[lo] == t) ? 1 : 0;
}

// D = A * (a_i a_j) off-diagonal, D_ii = A_ii always (mask diag is 1)
__global__ void build_D(const float* __restrict__ A, float* __restrict__ D,
                        const int* __restrict__ flags)
{
    const size_t i = (size_t)blockIdx.x * blockDim.x + threadIdx.x;
    const int r = (int)(i >> 11), c = (int)(i & 2047);
    const float v = A[i];
    D[i] = ((flags[r] & flags[c]) | (int)(r == c)) ? v : 0.0f;
}

__global__ void gather_coeff(const float* __restrict__ D, float* __restrict__ out,
                             const int* __restrict__ iv)
{
    const size_t i = (size_t)blockIdx.x * blockDim.x + threadIdx.x;  // 1M elems
    const int p = (int)(i >> 10), q = (int)(i & 1023);
    out[i] = D[(size_t)iv[p] * NDIM + iv[q]];
}

__global__ void gather_rows(const float* __restrict__ A, float* __restrict__ out,
                            const int* __restrict__ iv)
{
    const size_t i = (size_t)blockIdx.x * blockDim.x + threadIdx.x;  // 2M elems
    const int p = (int)(i >> 11), c = (int)(i & 2047);
    out[i] = A[(size_t)iv[p] * NDIM + c];
}

// ------------------------------------------------------------------
// FP32 WMMA GEMM: C = op(A) @ B, 2048 x 2048 x 2048, fp32 exact.
// Block: 256 threads (8 waves), 128x128 tile; wave = 64x32 (4x2 subtiles).
// K staged via LDS in 16-slabs, double buffered, k-major [16][136].
// B (and A when TRANSA) are staged with CDNA5 async memory->LDS copies
// (GLOBAL_LOAD_ASYNC_TO_LDS_B128, ASYNCcnt) -- no VGPR round trip; only
// the NN A operand needs a VGPR transpose-on-store.
// ------------------------------------------------------------------
template <bool TRANSA>
__global__ __launch_bounds__(256) void wmma_gemm(const float* __restrict__ A,
                                                 const float* __restrict__ B,
                                                 float* __restrict__ C)
{
    __shared__ alignas(16) float As[2][16][136];
    __shared__ alignas(16) float Bs[2][16][136];

    const int tid  = threadIdx.x;
    const int lane = tid & 31;
    const int wave = tid >> 5;
    const int wm = wave >> 2, wn = wave & 3;      // 2 x 4 wave grid
    const int half = lane >> 4, l15 = lane & 15;
    const int mBase = blockIdx.y * 128, nBase = blockIdx.x * 128;

    // loader index sets
    const int am  = tid & 127, akg = tid >> 7;    // NN A loader (transpose-on-store)
    const int tk  = tid >> 5,  tmg = tid & 31;    // async slab loader

    // async copy constants (GVS addressing: s[base] + v_off)
    const unsigned long long saddrB = (unsigned long long)(uintptr_t)(B + nBase);
    const unsigned long long saddrA = (unsigned long long)(uintptr_t)(A + mBase);
    const unsigned vstat     = (unsigned)((tk * NDIM + tmg * 4) * 4);  // byte offset
    const unsigned asBase    = (unsigned)(uintptr_t)(&As[0][0][0]);
    const unsigned bsBase    = (unsigned)(uintptr_t)(&Bs[0][0][0]);
    const unsigned ldsRow    = (unsigned)((tk * 136 + tmg * 4) * 4);
    const unsigned bufStride = 16u * 136u * 4u;
    const unsigned rowStr8   = 8u * 136u * 4u;
    const unsigned gStr8     = 8u * NDIM * 4u;
    const unsigned slabStr   = 16u * NDIM * 4u;

    const v8f zero = {0.f, 0.f, 0.f, 0.f, 0.f, 0.f, 0.f, 0.f};
    v8f acc[4][2];
#pragma unroll
    for (int i = 0; i < 4; ++i)
#pragma unroll
        for (int j = 0; j < 2; ++j) acc[i][j] = zero;

    float4 aR0, aR1;
    auto loadA_NN = [&](int kb) {     // 128 x 16 slab, contiguous in k
        const float* Ap = A + (size_t)(mBase + am) * NDIM + kb;
        aR0 = *reinterpret_cast<const float4*>(Ap + akg * 4);
        aR1 = *reinterpret_cast<const float4*>(Ap + (akg + 2) * 4);
    };
    auto storeA_NN = [&](int buf) {   // transpose-on-store: consecutive m -> no bank conflict
        As[buf][akg * 4 + 0][am] = aR0.x;  As[buf][akg * 4 + 1][am] = aR0.y;
        As[buf][akg * 4 + 2][am] = aR0.z;  As[buf][akg * 4 + 3][am] = aR0.w;
        As[buf][(akg + 2) * 4 + 0][am] = aR1.x;  As[buf][(akg + 2) * 4 + 1][am] = aR1.y;
        As[buf][(akg + 2) * 4 + 2][am] = aR1.z;  As[buf][(akg + 2) * 4 + 3][am] = aR1.w;
    };

    auto issueAsync = [&](int tslab, int buf) {
        const unsigned goff = vstat + (unsigned)tslab * slabStr;
        const unsigned lb   = bsBase + (unsigned)buf * bufStride + ldsRow;
        asm volatile("global_load_async_to_lds_b128 %0, %1, %2"
                     :: "v"(lb), "v"(goff), "s"(saddrB) : "memory");
        asm volatile("global_load_async_to_lds_b128 %0, %1, %2"
                     :: "v"(lb + rowStr8), "v"(goff + gStr8), "s"(saddrB) : "memory");
        if (TRANSA) {                 // op(A)[m,k] = A[k,m]: contiguous in m -> async too
            const unsigned la = asBase + (unsigned)buf * bufStride + ldsRow;
            asm volatile("global_load_async_to_lds_b128 %0, %1, %2"
                         :: "v"(la), "v"(goff), "s"(saddrA) : "memory");
            asm volatile("global_load_async_to_lds_b128 %0, %1, %2"
                         :: "v"(la + rowStr8), "v"(goff + gStr8), "s"(saddrA) : "memory");
        }
    };

    // prologue: stage slab 0 into buffer 0
    issueAsync(0, 0);
    if (!TRANSA) { loadA_NN(0); storeA_NN(0); }
    asm volatile("s_wait_asynccnt 0x0" ::: "memory");
    __syncthreads();

    const int NT = NDIM / 16;
    for (int t = 0; t < NT; ++t) {
        const int cur = t & 1;
        const bool pre = (t + 1 < NT);
        if (pre) {
            issueAsync(t + 1, cur ^ 1);
            if (!TRANSA) loadA_NN((t + 1) * 16);
        }

#pragma unroll
        for (int kq = 0; kq < 4; ++kq) {
            const int kk = kq * 4 + 2 * half;   // lanes 0-15: K=kk,kk+1; 16-31: kk+2,kk+3
            v2f a[4], b[2];
#pragma unroll
            for (int i = 0; i < 4; ++i) {
                const int m = wm * 64 + i * 16 + l15;
                a[i][0] = As[cur][kk][m];
                a[i][1] = As[cur][kk + 1][m];
            }
#pragma unroll
            for (int j = 0; j < 2; ++j) {
                const int n = wn * 32 + j * 16 + l15;
                b[j][0] = Bs[cur][kk][n];
                b[j][1] = Bs[cur][kk + 1][n];
            }
#pragma unroll
            for (int i = 0; i < 4; ++i)
#pragma unroll
                for (int j = 0; j < 2; ++j)
                    acc[i][j] = __builtin_amdgcn_wmma_f32_16x16x4_f32(
                        false, a[i], false, b[j], (short)0, acc[i][j], false, false);
        }

        if (pre) {
            if (!TRANSA) storeA_NN(cur ^ 1);
            asm volatile("s_wait_asynccnt 0x0" ::: "memory");
        }
        __syncthreads();
    }

    // C/D layout: vgpr r, lanes 0-15 -> row r; lanes 16-31 -> row r+8
#pragma unroll
    for (int i = 0; i < 4; ++i) {
#pragma unroll
        for (int j = 0; j < 2; ++j) {
            const int row0 = mBase + wm * 64 + i * 16 + half * 8;
            const int col  = nBase + wn * 32 + j * 16 + l15;
#pragma unroll
            for (int r = 0; r < 8; ++r)
                C[(size_t)(row0 + r) * NDIM + col] = acc[i][j][r];
        }
    }
}

// ------------------------------------------------------------------
// Launch
// ------------------------------------------------------------------
extern "C" void kernel_launch(void* const* d_in, const int* in_sizes, int n_in,
                              void* d_out, int out_size, void* d_ws, size_t ws_size,
                              hipStream_t stream)
{
    (void)in_sizes; (void)n_in; (void)out_size; (void)ws_size;

    const float* Ain  = (const float*)d_in[0];
    const float* O    = (const float*)d_in[1];
    const int*  sel   = (const int*)d_in[2];
    const int*  drop  = (const int*)d_in[3];
    const int*  act   = (const int*)d_in[4];
    const int*  inact = (const int*)d_in[5];

    const size_t NN = (size_t)NDIM * NDIM;   // 4194304

    float* out   = (float*)d_out;
    float* A_rec = out;                                   // 2048^2
    float* right = out + NN;                              // 2048^2
    float* D     = out + 2 * NN;                          // 2048^2
    float* mc    = out + 3 * NN;                          // 1024^2
    float* fc    = mc + (size_t)HALFN * HALFN;            // 1024^2
    float* mw    = fc + (size_t)HALFN * HALFN;            // 1024 x 2048
    float* fw    = mw + (size_t)HALFN * NDIM;             // 1024 x 2048

    float* Awork = (float*)d_ws;          // 16 MB
    float* T1    = Awork + NN;            // 16 MB
    int*   flags = (int*)(T1 + NN);       // 8 KB

    hipMemcpyAsync(Awork, Ain, NN * sizeof(float), hipMemcpyDeviceToDevice, stream);
    init_right<<<(int)(NN / 256), 256, 0, stream>>>(right);

    for (int l = 0; l < 256; ++l) {
        layer_rows<<<16, 256, 0, stream>>>(Awork, right, O, sel, l);
        layer_cols<<<8, 256, 0, stream>>>(Awork, mw, O, sel, drop, l);
    }

    make_flags<<<8, 256, 0, stream>>>(flags, act);
    build_D<<<(int)(NN / 256), 256, 0, stream>>>(Awork, D, flags);
    gather_coeff<<<(HALFN * HALFN) / 256, 256, 0, stream>>>(D, mc, inact);
    gather_coeff<<<(HALFN * HALFN) / 256, 256, 0, stream>>>(D, fc, act);
    gather_rows<<<(HALFN * NDIM) / 256, 256, 0, stream>>>(Awork, fw, act);

    dim3 g(NDIM / 128, NDIM / 128);
    wmma_gemm<false><<<g, 256, 0, stream>>>(D, right, T1);      // T1 = D @ right
    wmma_gemm<true ><<<g, 256, 0, stream>>>(right, T1, A_rec);  // A_rec = right^T @ T1
}